// DeepMOI_33114197852442
// MI455X (gfx1250) — compile-verified
//
#include <hip/hip_runtime.h>
#include <math.h>

// ---------------- problem constants ----------------
constexpr int N_GENES  = 20000;
constexpr int E_EDGES  = 200000;
constexpr int P_PATH   = 300;
constexpr int EP_EDGES = 2000;
constexpr int K_KEEP   = 16000;           // ceil(0.8 * 20000)
constexpr int TILES    = N_GENES / 16;    // 1250 (exact)

typedef float v2f __attribute__((ext_vector_type(2)));
typedef float v8f __attribute__((ext_vector_type(8)));

// D = A(16x4 f32) * B(4x16 f32) + C(16x16 f32), wave32 WMMA
__device__ __forceinline__ v8f wmma4(v2f a, v2f b, v8f c) {
  return __builtin_amdgcn_wmma_f32_16x16x4_f32(
      false, a, false, b, (short)0, c, false, false);
}

// A-matrix 16x4 layout: lanes 0-15 hold K=0 (v0), K=1 (v1); lanes 16-31 hold K=2 (v0), K=3 (v1)
__device__ __forceinline__ v2f make_a(float f0, float f1, float f2, int half) {
  v2f a; a.x = half ? f2 : f0; a.y = half ? 0.0f : f1; return a;
}
// B-matrix 4x16 from a row-major 3x3 weight (K padded to 4, N padded to 16)
// assumed layout mirrors C/D striping: v0 = rows K=0 (lanes 0-15) / K=2 (lanes 16-31); v1 = K=1 / K=3
__device__ __forceinline__ v2f make_b(const float* __restrict__ W, int n, int half) {
  int nc = (n < 3) ? n : 0;
  float w0 = W[(half ? 2 : 0) * 3 + nc];
  float w1 = W[1 * 3 + nc];
  v2f b;
  b.x = (n < 3) ? w0 : 0.0f;
  b.y = (n < 3 && !half) ? w1 : 0.0f;
  return b;
}

// order-preserving key: ascending unsigned order == ascending float order
__device__ __forceinline__ unsigned okey(float f) {
  unsigned u = __float_as_uint(f);
  return (u & 0x80000000u) ? ~u : (u | 0x80000000u);
}

// ---------------- utility kernels ----------------
__global__ void k_zero_f(float* p, long n) {
  long i = (long)blockIdx.x * blockDim.x + threadIdx.x;
  if (i < n) p[i] = 0.0f;
}
__global__ void k_state_init(unsigned* state) {
  int p = blockIdx.x * blockDim.x + threadIdx.x;
  if (p < P_PATH) { state[2 * p] = 0u; state[2 * p + 1] = (unsigned)K_KEEP; }
}

// ---------------- 1) hp = relu(x @ W_pool + b_pool); agg init = hp (self loops) ----------------
__global__ void k_hp(const float* __restrict__ x, const float* __restrict__ Wp,
                     const float* __restrict__ bp, float* __restrict__ hp,
                     unsigned* __restrict__ aggbits) {
  int wave = blockIdx.x * (blockDim.x >> 5) + (threadIdx.x >> 5);
  if (wave >= TILES) return;                       // wave-uniform
  int lane = threadIdx.x & 31, half = lane >> 4, m = lane & 15, n = m;
  int node = wave * 16 + m;
  float f0 = x[node * 3 + 0], f1 = x[node * 3 + 1], f2 = x[node * 3 + 2];
  v2f a = make_a(f0, f1, f2, half);
  v2f b = make_b(Wp, n, half);
  int nc = (n < 3) ? n : 0;
  float bv = bp[nc];
  float cv = (n < 3) ? bv : 0.0f;
  v8f c = {cv, cv, cv, cv, cv, cv, cv, cv};
  v8f d = wmma4(a, b, c);
  if (n < 3) {
#pragma unroll
    for (int j = 0; j < 8; ++j) {
      int M = j + 8 * half;
      int idx = (wave * 16 + M) * 3 + n;
      float v = fmaxf(d[j], 0.0f);
      hp[idx] = v;
      aggbits[idx] = __float_as_uint(v);           // non-negative: uint order == float order
    }
  }
}

// ---------------- 2) agg[dst] = max(agg[dst], hp[src]) over E global edges ----------------
__global__ void k_edge_max(const int* __restrict__ ei, const float* __restrict__ hp,
                           unsigned* __restrict__ aggbits) {
  int e = blockIdx.x * blockDim.x + threadIdx.x;
  if (e >= E_EDGES) return;
  int s = ei[e], d = ei[E_EDGES + e];
#pragma unroll
  for (int c = 0; c < 3; ++c)
    atomicMax(&aggbits[d * 3 + c], __float_as_uint(hp[s * 3 + c]));
}

// ---------------- 3) h = tanh(x@W_self + agg@W_neigh + b_conv) ----------------
__global__ void k_h(const float* __restrict__ x, const unsigned* __restrict__ aggbits,
                    const float* __restrict__ Ws, const float* __restrict__ Wn,
                    const float* __restrict__ bc, float* __restrict__ h) {
  int wave = blockIdx.x * (blockDim.x >> 5) + (threadIdx.x >> 5);
  if (wave >= TILES) return;
  int lane = threadIdx.x & 31, half = lane >> 4, m = lane & 15, n = m;
  int node = wave * 16 + m;
  float x0 = x[node * 3 + 0], x1 = x[node * 3 + 1], x2 = x[node * 3 + 2];
  float g0 = __uint_as_float(aggbits[node * 3 + 0]);
  float g1 = __uint_as_float(aggbits[node * 3 + 1]);
  float g2 = __uint_as_float(aggbits[node * 3 + 2]);
  v2f ax = make_a(x0, x1, x2, half);
  v2f ag = make_a(g0, g1, g2, half);
  v2f bs = make_b(Ws, n, half);
  v2f bn = make_b(Wn, n, half);
  int nc = (n < 3) ? n : 0;
  float bv = bc[nc];
  float cv = (n < 3) ? bv : 0.0f;
  v8f c = {cv, cv, cv, cv, cv, cv, cv, cv};
  v8f d1 = wmma4(ax, bs, c);
  v8f d2 = wmma4(ag, bn, d1);                      // chained accumulate
  if (n < 3) {
#pragma unroll
    for (int j = 0; j < 8; ++j) {
      int M = j + 8 * half;
      h[(wave * 16 + M) * 3 + n] = tanhf(d2[j]);
    }
  }
}

// ---------------- 4) pathway mean-aggregation scatter ----------------
__global__ void k_scatter_mean(const int* __restrict__ pe, const float* __restrict__ h,
                               float* __restrict__ msum, float* __restrict__ cnt) {
  long t = (long)blockIdx.x * blockDim.x + threadIdx.x;
  if (t >= (long)P_PATH * EP_EDGES) return;
  int p = (int)(t / EP_EDGES), i = (int)(t % EP_EDGES);
  const int* base = pe + (long)p * 2 * EP_EDGES;
  int s = base[i], d = base[EP_EDGES + i];
  long o = ((long)p * N_GENES + d) * 3;
#pragma unroll
  for (int c = 0; c < 3; ++c) atomicAdd(&msum[o + c], h[s * 3 + c]);
  atomicAdd(&cnt[(long)p * N_GENES + d], 1.0f);
}

// ---------------- 5) xc = relu(mean@Wl + bl + h@Wr), per pathway (WMMA); re-zero msum for asum ----------------
__global__ void k_xc(const float* __restrict__ h, float* __restrict__ msum,
                     const float* __restrict__ cnt,
                     const float* __restrict__ sub_Wl, const float* __restrict__ sub_bl,
                     const float* __restrict__ sub_Wr, float* __restrict__ xc) {
  int wave = blockIdx.x * (blockDim.x >> 5) + (threadIdx.x >> 5);  // exactly P*TILES waves
  int p = wave / TILES, t = wave % TILES;
  int lane = threadIdx.x & 31, half = lane >> 4, m = lane & 15, n = m;
  int node = t * 16 + m;
  long po = (long)p * N_GENES;
  float inv = 1.0f / fmaxf(cnt[po + node], 1.0f);
  long mo = (po + node) * 3;
  float m0 = msum[mo + 0] * inv, m1 = msum[mo + 1] * inv, m2 = msum[mo + 2] * inv;
  float h0 = h[node * 3 + 0], h1 = h[node * 3 + 1], h2 = h[node * 3 + 2];
  v2f am = make_a(m0, m1, m2, half);
  v2f ah = make_a(h0, h1, h2, half);
  const float* Wl = sub_Wl + p * 9;
  const float* Wr = sub_Wr + p * 9;
  const float* bl = sub_bl + p * 3;
  v2f bL = make_b(Wl, n, half);
  v2f bR = make_b(Wr, n, half);
  int nc = (n < 3) ? n : 0;
  float bv = bl[nc];
  float cv = (n < 3) ? bv : 0.0f;
  v8f c = {cv, cv, cv, cv, cv, cv, cv, cv};
  v8f d1 = wmma4(am, bL, c);
  v8f d2 = wmma4(ah, bR, d1);
  if (n < 3) {
#pragma unroll
    for (int j = 0; j < 8; ++j) {
      int M = j + 8 * half;
      long idx = (po + t * 16 + M) * 3 + n;
      xc[idx] = fmaxf(d2[j], 0.0f);
      msum[idx] = 0.0f;                            // recycle msum buffer as asum
    }
  }
}

// ---------------- 6) add-aggregation scatter of xc (into recycled buffer) ----------------
__global__ void k_scatter_add(const int* __restrict__ pe, const float* __restrict__ xc,
                              float* __restrict__ asum) {
  long t = (long)blockIdx.x * blockDim.x + threadIdx.x;
  if (t >= (long)P_PATH * EP_EDGES) return;
  int p = (int)(t / EP_EDGES), i = (int)(t % EP_EDGES);
  const int* base = pe + (long)p * 2 * EP_EDGES;
  int s = base[i], d = base[EP_EDGES + i];
  long os = ((long)p * N_GENES + s) * 3;
  long od = ((long)p * N_GENES + d) * 3;
#pragma unroll
  for (int c = 0; c < 3; ++c) atomicAdd(&asum[od + c], xc[os + c]);
}

// ---------------- 7) score = asum@Wrel + xc@Wroot + pb; histogram top byte ----------------
__global__ void k_score(const float* __restrict__ asum, const float* __restrict__ xc,
                        const float* __restrict__ Wrel, const float* __restrict__ Wroot,
                        const float* __restrict__ pb, float* __restrict__ score,
                        unsigned* __restrict__ hist) {
  long t = (long)blockIdx.x * blockDim.x + threadIdx.x;
  if (t >= (long)P_PATH * N_GENES) return;
  int p = (int)(t / N_GENES);
  long o = t * 3;
  float sc = asum[o] * Wrel[p * 3] + asum[o + 1] * Wrel[p * 3 + 1] + asum[o + 2] * Wrel[p * 3 + 2]
           + xc[o] * Wroot[p * 3] + xc[o + 1] * Wroot[p * 3 + 1] + xc[o + 2] * Wroot[p * 3 + 2]
           + pb[p];
  score[t] = sc;
  atomicAdd(&hist[p * 256 + (okey(sc) >> 24)], 1u);
}

// ---------------- 8) radix-select passes (find k-th largest score per pathway) ----------------
__global__ void k_hist_pass(const float* __restrict__ score, const unsigned* __restrict__ state,
                            unsigned* __restrict__ hist, int shift) {
  long t = (long)blockIdx.x * blockDim.x + threadIdx.x;
  if (t >= (long)P_PATH * N_GENES) return;
  int p = (int)(t / N_GENES);
  unsigned ou = okey(score[t]);
  unsigned himask = 0xFFFFFFFFu << (shift + 8);
  if ((ou & himask) == (state[2 * p] & himask))
    atomicAdd(&hist[p * 256 + ((ou >> shift) & 255u)], 1u);
}
__global__ void k_scan(unsigned* __restrict__ state, unsigned* __restrict__ hist, int shift) {
  int p = blockIdx.x * blockDim.x + threadIdx.x;
  if (p >= P_PATH) return;
  unsigned* hh = hist + p * 256;
  unsigned r = state[2 * p + 1];
  unsigned cum = 0; int chosen = 0;
  for (int b = 255; b >= 0; --b) {
    unsigned cb = hh[b];
    if (cum + cb >= r) { chosen = b; r = r - cum; break; }
    cum += cb;
  }
  state[2 * p] |= ((unsigned)chosen << shift);
  state[2 * p + 1] = r;                            // residual = #ties to include at final level
  for (int b = 0; b < 256; ++b) hh[b] = 0;         // clear for next pass
}

// ---------------- 9) gated softmax readout over selected nodes (one block per pathway) ----------------
__global__ void __launch_bounds__(256)
k_readout(const float* __restrict__ score, const float* __restrict__ xc,
          const unsigned* __restrict__ state,
          const float* __restrict__ gW, const float* __restrict__ gb,
          float* __restrict__ readout) {
  int p = blockIdx.x;
  int tid = threadIdx.x, lane = tid & 31, wid = tid >> 5;
  unsigned T = state[2 * p];
  unsigned tneed = state[2 * p + 1];
  float g0 = gW[p * 3], g1 = gW[p * 3 + 1], g2 = gW[p * 3 + 2], gbv = gb[p];
  long po = (long)p * N_GENES;
  __shared__ unsigned wcnt[8];
  __shared__ float red[256];
  float accw = 0.0f, a0 = 0.0f, a1 = 0.0f, a2 = 0.0f;
  unsigned tiebase = 0;
  for (int base = 0; base < N_GENES; base += 256) {
    int i = base + tid;
    bool valid = i < N_GENES;
    float sc = valid ? score[po + i] : 0.0f;
    if (valid) {
      __builtin_prefetch(&score[po + i + 256], 0, 0);
      __builtin_prefetch(&xc[(po + i + 256) * 3], 0, 0);
    }
    unsigned ou = okey(sc);
    bool gt  = valid && (ou > T);
    bool tie = valid && (ou == T);
    unsigned mask = (unsigned)__ballot(tie);       // wave32: low 32 bits
    unsigned lanepre = __popc(mask & ((1u << lane) - 1u));
    __syncthreads();
    if (lane == 0) wcnt[wid] = __popc(mask);
    __syncthreads();
    unsigned woff = 0, btot = 0;
#pragma unroll
    for (int w = 0; w < 8; ++w) { unsigned cw = wcnt[w]; btot += cw; if (w < wid) woff += cw; }
    bool sel = gt || (tie && (tiebase + woff + lanepre) < tneed);  // index-ordered tie break
    tiebase += btot;
    if (sel) {
      float th = tanhf(sc);
      long o = (po + i) * 3;
      float xk0 = xc[o] * th, xk1 = xc[o + 1] * th, xk2 = xc[o + 2] * th;
      float li = xk0 * g0 + xk1 * g1 + xk2 * g2 + gbv;
      float w = expf(li);                          // logits ~1e-3: no overflow, exact softmax
      accw += w; a0 += w * xk0; a1 += w * xk1; a2 += w * xk2;
    }
  }
  float vals[4] = {accw, a0, a1, a2};
  float out[4];
#pragma unroll
  for (int v = 0; v < 4; ++v) {
    __syncthreads();
    red[tid] = vals[v];
    __syncthreads();
    for (int s = 128; s > 0; s >>= 1) { if (tid < s) red[tid] += red[tid + s]; __syncthreads(); }
    out[v] = red[0];
  }
  if (tid == 0) {
    float inv = 1.0f / out[0];
    readout[p * 3 + 0] = fmaxf(out[1] * inv, 0.0f);
    readout[p * 3 + 1] = fmaxf(out[2] * inv, 0.0f);
    readout[p * 3 + 2] = fmaxf(out[3] * inv, 0.0f);
  }
}

// ---------------- 10) final MLP + sigmoid ----------------
__global__ void k_final(const float* __restrict__ readout, const float* __restrict__ linW,
                        const float* __restrict__ linb, const float* __restrict__ mlpW,
                        const float* __restrict__ mlpb, float* __restrict__ out) {
  __shared__ float red[512];
  int tid = threadIdx.x;
  float v = 0.0f;
  if (tid < P_PATH) {
    float r = readout[tid * 3] * linW[0] + readout[tid * 3 + 1] * linW[1]
            + readout[tid * 3 + 2] * linW[2] + linb[0];
    v = fmaxf(r, 0.0f) * mlpW[tid];                // relu(relu(x)) == relu(x)
  }
  red[tid] = v; __syncthreads();
  for (int s = 256; s > 0; s >>= 1) { if (tid < s) red[tid] += red[tid + s]; __syncthreads(); }
  if (tid == 0) out[0] = 1.0f / (1.0f + expf(-(red[0] + mlpb[0])));
}

// ---------------- launch ----------------
extern "C" void kernel_launch(void* const* d_in, const int* in_sizes, int n_in,
                              void* d_out, int out_size, void* d_ws, size_t ws_size,
                              hipStream_t stream) {
  (void)in_sizes; (void)n_in; (void)out_size; (void)ws_size;
  const float* x       = (const float*)d_in[0];
  const int*   ei      = (const int*)d_in[1];
  const int*   pe      = (const int*)d_in[2];
  const float* W_pool  = (const float*)d_in[3];
  const float* b_pool  = (const float*)d_in[4];
  const float* W_self  = (const float*)d_in[5];
  const float* W_neigh = (const float*)d_in[6];
  const float* b_conv  = (const float*)d_in[7];
  const float* sub_Wl  = (const float*)d_in[8];
  const float* sub_bl  = (const float*)d_in[9];
  const float* sub_Wr  = (const float*)d_in[10];
  const float* Wrel    = (const float*)d_in[11];
  const float* Wroot   = (const float*)d_in[12];
  const float* pb      = (const float*)d_in[13];
  const float* gW      = (const float*)d_in[14];
  const float* gb      = (const float*)d_in[15];
  const float* linW    = (const float*)d_in[16];
  const float* linb    = (const float*)d_in[17];
  const float* mlpW    = (const float*)d_in[18];
  const float* mlpb    = (const float*)d_in[19];
  float* out = (float*)d_out;

  // workspace layout (floats)
  float*    ws      = (float*)d_ws;
  float*    hp      = ws;                                   // 60,000
  unsigned* aggbits = (unsigned*)(ws + 60000);              // 60,000
  float*    h       = ws + 120000;                          // 60,000
  float*    msum    = ws + 180000;                          // 18,000,000 (recycled as asum)
  float*    cnt     = msum + (size_t)18000000;              // 6,000,000
  float*    xc      = cnt + (size_t)6000000;                // 18,000,000
  float*    score   = xc + (size_t)18000000;                // 6,000,000
  unsigned* hist    = (unsigned*)(score + (size_t)6000000); // 76,800
  unsigned* state   = hist + 76800;                         // 600
  float*    readout = (float*)(state + 600);                // 900

  // init (msum+cnt contiguous)
  { long n = 24000000; k_zero_f<<<(unsigned)((n + 255) / 256), 256, 0, stream>>>(msum, n); }
  { long n = 76800;    k_zero_f<<<(unsigned)((n + 255) / 256), 256, 0, stream>>>((float*)hist, n); }
  k_state_init<<<2, 256, 0, stream>>>(state);

  const int WPB = 8;  // waves per 256-thread block
  k_hp<<<(TILES + WPB - 1) / WPB, 256, 0, stream>>>(x, W_pool, b_pool, hp, aggbits);
  k_edge_max<<<(E_EDGES + 255) / 256, 256, 0, stream>>>(ei, hp, aggbits);
  k_h<<<(TILES + WPB - 1) / WPB, 256, 0, stream>>>(x, aggbits, W_self, W_neigh, b_conv, h);

  { long n = (long)P_PATH * EP_EDGES;
    k_scatter_mean<<<(unsigned)((n + 255) / 256), 256, 0, stream>>>(pe, h, msum, cnt); }
  k_xc<<<(P_PATH * TILES) / WPB, 256, 0, stream>>>(h, msum, cnt, sub_Wl, sub_bl, sub_Wr, xc);
  { long n = (long)P_PATH * EP_EDGES;
    k_scatter_add<<<(unsigned)((n + 255) / 256), 256, 0, stream>>>(pe, xc, msum); }

  { long n = (long)P_PATH * N_GENES;
    unsigned g = (unsigned)((n + 255) / 256);
    k_score<<<g, 256, 0, stream>>>(msum, xc, Wrel, Wroot, pb, score, hist);
    k_scan<<<2, 256, 0, stream>>>(state, hist, 24);
    k_hist_pass<<<g, 256, 0, stream>>>(score, state, hist, 16);
    k_scan<<<2, 256, 0, stream>>>(state, hist, 16);
    k_hist_pass<<<g, 256, 0, stream>>>(score, state, hist, 8);
    k_scan<<<2, 256, 0, stream>>>(state, hist, 8);
    k_hist_pass<<<g, 256, 0, stream>>>(score, state, hist, 0);
    k_scan<<<2, 256, 0, stream>>>(state, hist, 0);
  }

  k_readout<<<P_PATH, 256, 0, stream>>>(score, xc, state, gW, gb, readout);
  k_final<<<1, 512, 0, stream>>>(readout, linW, linb, mlpW, mlpb, out);
}